// Model_14980845928514
// MI455X (gfx1250) — compile-verified
//
#include <hip/hip_runtime.h>

#define NN 50000
#define EE 600000
#define RR 32
#define BB 8

typedef __attribute__((ext_vector_type(2))) float v2f;
typedef __attribute__((ext_vector_type(8))) float v8f;

__device__ __forceinline__ void atomAddF(float* p, float v) {
    __hip_atomic_fetch_add(p, v, __ATOMIC_RELAXED, __HIP_MEMORY_SCOPE_AGENT);
}

// Low 32 bits of a flat shared-aperture address are the wave-relative LDS byte
// address (ISA 10.2).
__device__ __forceinline__ unsigned ldsOffsetOf(const void* p) {
    return (unsigned)(size_t)p;
}

// CDNA5 async DMA: global -> LDS, 16 bytes, tracked by ASYNCcnt.
__device__ __forceinline__ void asyncLoad16(unsigned ldsOff, const void* g) {
    asm volatile("global_load_async_to_lds_b128 %0, %1, off"
                 :: "v"(ldsOff), "v"(g) : "memory");
}
__device__ __forceinline__ void waitAsync() {
    asm volatile("s_wait_asynccnt 0x0" ::: "memory");
}

// ---------------- segment counts -> reciprocals ----------------
__global__ void zero_u32_k(unsigned* p, int n) {
    int i = blockIdx.x * blockDim.x + threadIdx.x;
    if (i < n) p[i] = 0u;
}

__global__ void count_edges_k(const int* __restrict__ dst, const int* __restrict__ et,
                              unsigned* __restrict__ cnt) {
    int e = blockIdx.x * blockDim.x + threadIdx.x;
    if (e < EE) {
        unsigned seg = (unsigned)et[e] * NN + (unsigned)dst[e];
        atomicAdd(&cnt[seg], 1u);
    }
}

__global__ void make_rcp_k(unsigned* buf, int n) {
    int i = blockIdx.x * blockDim.x + threadIdx.x;
    if (i < n) {
        float c = (float)buf[i];
        buf[i] = __float_as_uint(1.0f / fmaxf(c, 1.0f));
    }
}

// ---------------- fp32 WMMA GEMM: C[N,Dout] = A[N,128] x W[128,Dout] (+bias) ----------------
// blockDim = 256 (8 waves); each block: 16 rows x 128 cols. grid.y picks col slice.
// A staged via async-DMA to LDS; W slice staged transposed (ldsW[c][k], stride 133 -> both
// lane-halves' ds_load_b64 feeds are bank-conflict free). Dout is compile-time so the
// epilogue stores use immediate offsets and staging avoids 64-bit multiplies.
template <int DOUT>
__global__ void gemm_wmma_f32(const float* __restrict__ A, const float* __restrict__ W,
                              const float* __restrict__ bias, float* __restrict__ C,
                              int colTiles) {
    __shared__ __align__(16) float ldsA[16 * 132];
    __shared__ __align__(16) float ldsW[128 * 133];
    const int rowTile = blockIdx.x;
    const int nWaves  = blockDim.x >> 5;            // 8
    const int wave    = threadIdx.x >> 5;
    const int lane    = threadIdx.x & 31;
    const int tid     = threadIdx.x;
    const int colBase = blockIdx.y * (nWaves << 4); // 128 cols per block

    // --- async stage of the 16x128 A tile in 16B chunks (ASYNCcnt path) ---
    const float* arow = A + (size_t)rowTile * 16 * 128;
    for (int i = tid; i < 512; i += blockDim.x) {
        int r = i >> 5, c16 = i & 31;
        asyncLoad16(ldsOffsetOf(&ldsA[r * 132 + c16 * 4]), arow + r * 128 + c16 * 4);
    }
    // --- stage W slice transposed while the async DMA runs ---
    {
        int c  = tid & 127;
        int k0 = tid >> 7;                 // 0 or 1
        int gc = colBase + c;
        for (int k = k0; k < 128; k += 2) {
            float v = (gc < DOUT) ? W[k * DOUT + gc] : 0.0f;
            ldsW[c * 133 + k] = v;
        }
    }
    waitAsync();
    __syncthreads();

    const int colTile = blockIdx.y * nWaves + wave;
    if (colTile >= colTiles) return;       // wave-uniform: EXEC all-ones for WMMA

    const int m  = lane & 15;              // M for A, N for B/C/D
    const int kg = (lane >> 4) << 1;       // K sub-offset 0 / 2 per lane half
    const float* aRow = &ldsA[m * 132 + kg];
    const float* wRow = &ldsW[(wave * 16 + m) * 133 + kg];

    v8f acc = {};
#pragma unroll 4
    for (int k = 0; k < 128; k += 4) {
        v2f a, b;
        a.x = aRow[k];
        a.y = aRow[k + 1];
        b.x = wRow[k];
        b.y = wRow[k + 1];
        acc = __builtin_amdgcn_wmma_f32_16x16x4_f32(false, a, false, b,
                                                    (short)0, acc, false, false);
    }

    const int col   = colTile * 16 + m;
    const float bv  = bias ? bias[col] : 0.0f;
    const int rbase = rowTile * 16 + ((lane >> 4) << 3);
    float* cp = C + (size_t)rbase * DOUT + col;
#pragma unroll
    for (int v = 0; v < 8; v++) {
        cp[v * DOUT] = acc[v] + bv;        // immediate byte offsets (v*DOUT*4 < 2^23)
    }
}

// -------- build Wcat1 chunk [128 x 512] = [V1_{4c} | V1_{4c+1} | V1_{4c+2} | V1_{4c+3}] ------
__global__ void build_w1_k(const float* __restrict__ V1, float* __restrict__ w1, int chunk) {
    int i = blockIdx.x * blockDim.x + threadIdx.x;
    if (i < 128 * 512) {
        int k = i >> 9, c = i & 511;
        int b = c >> 7, o = c & 127;
        w1[i] = V1[(size_t)(chunk * 4 + b) * 128 * 128 + k * 128 + o];
    }
}

// ---------------- layer-1 edge scatter: 4 bases fused per pass, one wave per edge ------------
__global__ void edge_pass1_k(const float* __restrict__ proj, const int* __restrict__ ei,
                             const int* __restrict__ et, const float* __restrict__ rcp,
                             const float* __restrict__ a1, int chunk, float* __restrict__ z1) {
    long long t = (long long)blockIdx.x * blockDim.x + threadIdx.x;
    int e = (int)(t >> 5);
    if (e >= EE) return;
    int lane = (int)(t & 31);
    int s = ei[e], d = ei[EE + e], r = et[e];
    float inv = rcp[(size_t)r * NN + d];
    const float* ar = a1 + r * BB + chunk * 4;
    float c0 = ar[0] * inv, c1 = ar[1] * inv, c2 = ar[2] * inv, c3 = ar[3] * inv;
    const float4* pp = (const float4*)(proj + (size_t)s * 512);
    float4 p0 = pp[lane], p1 = pp[32 + lane], p2 = pp[64 + lane], p3 = pp[96 + lane];
    float vx = c0 * p0.x + c1 * p1.x + c2 * p2.x + c3 * p3.x;
    float vy = c0 * p0.y + c1 * p1.y + c2 * p2.y + c3 * p3.y;
    float vz = c0 * p0.z + c1 * p1.z + c2 * p2.z + c3 * p3.z;
    float vw = c0 * p0.w + c1 * p1.w + c2 * p2.w + c3 * p3.w;
    float* zp = z1 + (size_t)d * 128 + lane * 4;
    atomAddF(zp + 0, vx);
    atomAddF(zp + 1, vy);
    atomAddF(zp + 2, vz);
    atomAddF(zp + 3, vw);
}

__global__ void relu_k(float* p, int n) {
    int i = blockIdx.x * blockDim.x + threadIdx.x;
    if (i < n) p[i] = fmaxf(p[i], 0.0f);
}

// ---------------- build Wcat2 [128 x 288] = [V2_0 .. V2_7 | root2] ----------------
__global__ void build_w2_k(const float* __restrict__ V2, const float* __restrict__ root2,
                           float* __restrict__ w2) {
    int i = blockIdx.x * blockDim.x + threadIdx.x;
    if (i < 128 * 288) {
        int k = i / 288, c = i % 288;
        w2[i] = (c < 256) ? V2[(size_t)(c >> 5) * 128 * 32 + k * 32 + (c & 31)]
                          : root2[k * 32 + (c - 256)];
    }
}

__global__ void init_out_k(const float* __restrict__ hp2, const float* __restrict__ bias2,
                           float* __restrict__ out) {
    int i = blockIdx.x * blockDim.x + threadIdx.x;
    if (i < NN * 32) {
        int n = i >> 5, o = i & 31;
        out[i] = hp2[(size_t)n * 288 + 256 + o] + bias2[o];
    }
}

// ---------------- layer-2 edge scatter: lane = class, all 8 bases fused in registers ---------
__global__ void edge_pass2_k(const float* __restrict__ hp2, const int* __restrict__ ei,
                             const int* __restrict__ et, const float* __restrict__ rcp,
                             const float* __restrict__ a2, float* __restrict__ out) {
    long long t = (long long)blockIdx.x * blockDim.x + threadIdx.x;
    int e = (int)(t >> 5);
    if (e >= EE) return;
    int lane = (int)(t & 31);
    int s = ei[e], d = ei[EE + e], r = et[e];
    float inv = rcp[(size_t)r * NN + d];
    const float* base = hp2 + (size_t)s * 288;
    float sum = 0.0f;
#pragma unroll
    for (int b = 0; b < 8; b++) sum += a2[r * BB + b] * base[b * 32 + lane];
    atomAddF(&out[(size_t)d * 32 + lane], inv * sum);
}

// ---------------- softmax over 32 classes: one wave per row ----------------
__global__ void softmax32_k(float* out) {
    long long t = (long long)blockIdx.x * blockDim.x + threadIdx.x;
    int row = (int)(t >> 5);
    if (row >= NN) return;
    int lane = (int)(t & 31);
    float v = out[(size_t)row * 32 + lane];
    float m = v;
    for (int off = 16; off; off >>= 1) m = fmaxf(m, __shfl_xor(m, off, 32));
    float ex = __expf(v - m);
    float s = ex;
    for (int off = 16; off; off >>= 1) s += __shfl_xor(s, off, 32);
    out[(size_t)row * 32 + lane] = ex / s;
}

extern "C" void kernel_launch(void* const* d_in, const int* in_sizes, int n_in,
                              void* d_out, int out_size, void* d_ws, size_t ws_size,
                              hipStream_t stream) {
    const float* x     = (const float*)d_in[0];
    const float* V1    = (const float*)d_in[1];
    const float* a1    = (const float*)d_in[2];
    const float* root1 = (const float*)d_in[3];
    const float* bias1 = (const float*)d_in[4];
    const float* V2    = (const float*)d_in[5];
    const float* a2    = (const float*)d_in[6];
    const float* root2 = (const float*)d_in[7];
    const float* bias2 = (const float*)d_in[8];
    const int*   ei    = (const int*)d_in[9];    // [2,E]: src row 0, dst row 1
    const int*   et    = (const int*)d_in[10];   // [E]
    float* out = (float*)d_out;

    float* ws    = (float*)d_ws;
    float* rcp   = ws;                             // RR*NN       ( 6.4 MB)
    float* z1    = rcp   + (size_t)RR * NN;        // NN*128      (25.6 MB)
    float* proj  = z1    + (size_t)NN * 128;       // NN*512      (102.4 MB)
    float* hp2   = proj  + (size_t)NN * 512;       // NN*288      (57.6 MB)
    float* w2    = hp2   + (size_t)NN * 288;       // 128*288
    float* wcat1 = w2    + (size_t)128 * 288;      // 128*512

    const int rn = RR * NN;
    zero_u32_k   <<<(rn + 255) / 256, 256, 0, stream>>>((unsigned*)rcp, rn);
    count_edges_k<<<(EE + 255) / 256, 256, 0, stream>>>(ei + EE, et, (unsigned*)rcp);
    make_rcp_k   <<<(rn + 255) / 256, 256, 0, stream>>>((unsigned*)rcp, rn);

    // layer 1: z1 = x @ root1 + bias1, then 2 chunks of (4-basis projection + fused scatter)
    gemm_wmma_f32<128><<<dim3(NN / 16, 1), 256, 0, stream>>>(x, root1, bias1, z1, 8);
    for (int chunk = 0; chunk < 2; chunk++) {
        build_w1_k        <<<(128 * 512 + 255) / 256, 256, 0, stream>>>(V1, wcat1, chunk);
        gemm_wmma_f32<512><<<dim3(NN / 16, 4), 256, 0, stream>>>(x, wcat1, nullptr, proj, 32);
        edge_pass1_k      <<<(EE * 32) / 256, 256, 0, stream>>>(proj, ei, et, rcp, a1, chunk, z1);
    }
    relu_k<<<(NN * 128 + 255) / 256, 256, 0, stream>>>(z1, NN * 128);

    // layer 2: hp2 = relu(z1) @ [V2 bases | root2], then fused basis-sum scatter
    build_w2_k        <<<(128 * 288 + 255) / 256, 256, 0, stream>>>(V2, root2, w2);
    gemm_wmma_f32<288><<<dim3(NN / 16, 3), 256, 0, stream>>>(z1, w2, nullptr, hp2, 18);
    init_out_k        <<<(NN * 32 + 255) / 256, 256, 0, stream>>>(hp2, bias2, out);
    edge_pass2_k      <<<(EE * 32) / 256, 256, 0, stream>>>(hp2, ei, et, rcp, a2, out);
    softmax32_k       <<<(NN * 32 + 255) / 256, 256, 0, stream>>>(out);
}